// DecoderWithAttention_82789789597980
// MI455X (gfx1250) — compile-verified
//
#include <hip/hip_runtime.h>
#include <math.h>

#define B_   64
#define P_   196       // 14*14
#define ENC_ 2048
#define A_   512
#define E_   512
#define D_   512
#define V_   10000
#define T_   21
#define TDEC_ 20

typedef float v2f __attribute__((ext_vector_type(2)));
typedef float v8f __attribute__((ext_vector_type(8)));

__device__ __forceinline__ float sigmoidf_(float x){ return 1.0f/(1.0f+__expf(-x)); }

// ---------------------------------------------------------------------------
// fp32 WMMA GEMM: C[M,N] = act(A[M,K] @ B[K,N] + bias), optional BT (B stored
// [N,K] row-major -> A @ B^T). Each wave computes an MT*16 x 16 output strip:
// the B fragment is loaded ONCE per k-step and reused across MT back-to-back
// WMMAs (4x less B-operand L2 traffic; B is the dominant traffic since M=64).
// 4 waves per block along N. K stepped by 4 via V_WMMA_F32_16X16X4_F32.
// Fragment layouts per CDNA5 ISA 7.12.2:
//   A 16x4 : lane l -> M = l&15, K = (l>>4)*2 + {0,1} in vgpr {0,1}
//   B 4x16 : lane l -> N = l&15, K = (l>>4)*2 + {0,1} in vgpr {0,1}
//   C 16x16: lane l -> N = l&15, M = vgpr + 8*(l>>4)
// ---------------------------------------------------------------------------
template<int ACT, bool BT, bool GATHER, bool ACCUM, bool MASK, int MT>
__global__ __launch_bounds__(128) void wmma_gemm(
    const float* __restrict__ Am, int lda,
    const float* __restrict__ Bm, int ldb,
    const float* __restrict__ bias,
    float* __restrict__ Cm, int ldc,
    int N, int K,
    const int* __restrict__ rowidx,
    const int* __restrict__ dlen, int t)
{
  const int wave = threadIdx.x >> 5;
  const int lane = threadIdx.x & 31;
  const int tileN = blockIdx.x * 4 + wave;
  if (tileN * 16 >= N) return;             // whole-wave uniform exit (EXEC all-1 for WMMA)
  const int tileM0 = blockIdx.y * MT;
  const int half = lane >> 4;
  const int lidx = lane & 15;
  const int col = tileN * 16 + lidx;       // B/C column for this lane

  const float* __restrict__ Aptr[MT];
  #pragma unroll
  for (int m = 0; m < MT; m++) {
    const int row = (tileM0 + m) * 16 + lidx;
    const long arow = GATHER ? (long)rowidx[row] : (long)row;
    Aptr[m] = Am + arow * (long)lda;
  }

  v8f acc[MT];
  #pragma unroll
  for (int m = 0; m < MT; m++) {
    if (ACCUM) {
      #pragma unroll
      for (int i = 0; i < 8; i++)
        acc[m][i] = Cm[(long)((tileM0 + m)*16 + i + half*8) * ldc + col];
    } else {
      const float bv = bias ? bias[col] : 0.0f;
      #pragma unroll
      for (int i = 0; i < 8; i++) acc[m][i] = bv;
    }
  }

  #pragma unroll 2
  for (int k = 0; k < K; k += 4) {
    v2f b;
    if (BT) {   // B stored [N,K]: B^T[k][n] = Bm[n*ldb + k]
      b.x = Bm[(long)col * ldb + k + half*2];
      b.y = Bm[(long)col * ldb + k + half*2 + 1];
    } else {    // B stored [K,N]
      b.x = Bm[(long)(k + half*2)     * ldb + col];
      b.y = Bm[(long)(k + half*2 + 1) * ldb + col];
    }
    #pragma unroll
    for (int m = 0; m < MT; m++) {
      v2f a;
      a.x = Aptr[m][k + half*2];
      a.y = Aptr[m][k + half*2 + 1];
      acc[m] = __builtin_amdgcn_wmma_f32_16x16x4_f32(false, a, false, b, (short)0,
                                                     acc[m], false, false);
    }
  }

  #pragma unroll
  for (int m = 0; m < MT; m++) {
    #pragma unroll
    for (int i = 0; i < 8; i++) {
      const int r = (tileM0 + m)*16 + i + half*8;
      float v = acc[m][i];
      if (ACT == 1) v = tanhf(v);
      else if (ACT == 2) v = sigmoidf_(v);
      if (MASK) { if (!(t < dlen[r])) v = 0.0f; }
      Cm[(long)r * ldc + col] = v;
    }
  }
}

// ---------------------------------------------------------------------------
// Stable descending argsort of clipped caption lengths (B=64, O(B^2) ranks),
// gather captions, emit int scratch + float outputs (caps, dec_len, sort_ind).
// ---------------------------------------------------------------------------
__global__ void prep_kernel(const int* __restrict__ cap_len,
                            const int* __restrict__ captions,
                            int* __restrict__ sort_ind, int* __restrict__ dec_len,
                            int* __restrict__ caps_s,
                            float* __restrict__ out_caps, float* __restrict__ out_declen,
                            float* __restrict__ out_sortind)
{
  __shared__ int cl[B_];
  const int i = threadIdx.x;
  int v = cap_len[i]; if (v < 1) v = 1;
  cl[i] = v;
  __syncthreads();
  int rank = 0;
  for (int j = 0; j < B_; j++)
    if (cl[j] > v || (cl[j] == v && j < i)) rank++;
  sort_ind[rank]    = i;
  dec_len[rank]     = v - 1;
  out_sortind[rank] = (float)i;
  out_declen[rank]  = (float)(v - 1);
  for (int tt = 0; tt < T_; tt++) {
    const int cv = captions[i*T_ + tt];
    caps_s[rank*T_ + tt]   = cv;
    out_caps[rank*T_ + tt] = (float)cv;
  }
}

__global__ void rowidx_kernel(const int* __restrict__ sort_ind, int* __restrict__ rowidx) {
  const int r = blockIdx.x * blockDim.x + threadIdx.x;
  if (r >= B_*P_) return;
  rowidx[r] = sort_ind[r / P_] * P_ + (r % P_);
}

__global__ void mean_kernel(const float* __restrict__ enc, const int* __restrict__ sort_ind,
                            float* __restrict__ mean_enc) {
  const int b = blockIdx.x;
  const int base = sort_ind[b] * P_;
  for (int ch = threadIdx.x; ch < ENC_; ch += blockDim.x) {
    float s = 0.f;
    for (int p = 0; p < P_; p++) s += enc[(long)(base + p) * ENC_ + ch];
    mean_enc[b*ENC_ + ch] = s * (1.0f / P_);
  }
}

// e[b,p] = sum_a relu(att1[b,p,a] + att2[b,a]) * fullW[a] + fullb  (one wave per (b,p))
__global__ __launch_bounds__(256) void e_kernel(const float* __restrict__ att1,
                                                const float* __restrict__ att2,
                                                const float* __restrict__ fullW,
                                                const float* __restrict__ fullb,
                                                float* __restrict__ e_buf)
{
  const int wave = threadIdx.x >> 5;
  const int lane = threadIdx.x & 31;
  const int idx = blockIdx.x * 8 + wave;      // b*P + p  (12544 total, grid 1568)
  const int b = idx / P_;
  const float* __restrict__ a1 = att1 + (long)idx * A_;
  const float* __restrict__ a2 = att2 + b * A_;
  float s = 0.f;
  for (int a = lane; a < A_; a += 32) {
    float v = a1[a] + a2[a];
    v = v > 0.f ? v : 0.f;
    s += v * fullW[a];
  }
  #pragma unroll
  for (int off = 16; off; off >>= 1) s += __shfl_xor(s, off, 32);
  if (lane == 0) e_buf[idx] = s + fullb[0];
}

// softmax over P, write (masked) alphas to d_out, then awe[b,:] = sum_p alpha*enc
__global__ __launch_bounds__(256) void softmax_awe_kernel(
    const float* __restrict__ e_buf, const float* __restrict__ enc,
    const int* __restrict__ sort_ind, const int* __restrict__ dec_len, int t,
    float* __restrict__ awe, float* __restrict__ out_alphas)
{
  __shared__ float sh[P_];
  __shared__ float red[256];
  const int b = blockIdx.x;
  const int tid = threadIdx.x;
  if (tid < P_) sh[tid] = e_buf[b*P_ + tid];
  __syncthreads();
  float m = -1e30f;
  if (tid < P_) m = sh[tid];
  red[tid] = m; __syncthreads();
  for (int s = 128; s; s >>= 1) { if (tid < s) red[tid] = fmaxf(red[tid], red[tid+s]); __syncthreads(); }
  m = red[0]; __syncthreads();
  float sum = (tid < P_) ? __expf(sh[tid] - m) : 0.f;
  red[tid] = sum; __syncthreads();
  for (int s = 128; s; s >>= 1) { if (tid < s) red[tid] += red[tid+s]; __syncthreads(); }
  const float inv = 1.0f / red[0];
  __syncthreads();
  const bool active = t < dec_len[b];
  if (tid < P_) {
    const float al = __expf(sh[tid] - m) * inv;
    out_alphas[(long)b*TDEC_*P_ + (long)t*P_ + tid] = active ? al : 0.0f;
    sh[tid] = al;
  }
  __syncthreads();
  const int base = sort_ind[b] * P_;
  for (int ch = tid; ch < ENC_; ch += 256) {
    float s = 0.f;
    for (int p = 0; p < P_; p++) s += sh[p] * enc[(long)(base + p) * ENC_ + ch];
    awe[b*ENC_ + ch] = s;
  }
}

// x = [ emb[token] ; gate * awe ]
__global__ void build_x_kernel(const float* __restrict__ emb, const int* __restrict__ caps_s, int t,
                               const float* __restrict__ gate, const float* __restrict__ awe,
                               float* __restrict__ x)
{
  const int b = blockIdx.x;
  const long tok = caps_s[b*T_ + t];
  for (int j = threadIdx.x; j < E_ + ENC_; j += blockDim.x) {
    float v;
    if (j < E_) v = emb[tok * E_ + j];
    else        v = gate[b*ENC_ + (j - E_)] * awe[b*ENC_ + (j - E_)];
    x[b*(E_+ENC_) + j] = v;
  }
}

__global__ void lstm_kernel(const float* __restrict__ gates, const float* __restrict__ b_hh,
                            const int* __restrict__ dec_len, int t,
                            float* __restrict__ h, float* __restrict__ c,
                            float* __restrict__ h_new)
{
  const int b = blockIdx.x;
  const bool active = t < dec_len[b];
  for (int d = threadIdx.x; d < D_; d += blockDim.x) {
    const float gi = gates[b*4*D_ +          d] + b_hh[         d];
    const float gf = gates[b*4*D_ +   D_  +  d] + b_hh[  D_  +  d];
    const float gg = gates[b*4*D_ + 2*D_  +  d] + b_hh[2*D_  +  d];
    const float go = gates[b*4*D_ + 3*D_  +  d] + b_hh[3*D_  +  d];
    const float cn = sigmoidf_(gf) * c[b*D_ + d] + sigmoidf_(gi) * tanhf(gg);
    const float hn = sigmoidf_(go) * tanhf(cn);
    h_new[b*D_ + d] = hn;
    if (active) { h[b*D_ + d] = hn; c[b*D_ + d] = cn; }
  }
}

// ---------------------------------------------------------------------------
extern "C" void kernel_launch(void* const* d_in, const int* in_sizes, int n_in,
                              void* d_out, int out_size, void* d_ws, size_t ws_size,
                              hipStream_t stream) {
  const float* enc       = (const float*)d_in[0];   // (B,14,14,ENC) == (B,P,ENC)
  const int*   captions  = (const int*)  d_in[1];   // (B,T)
  const int*   cap_len   = (const int*)  d_in[2];   // (B,)
  const float* emb       = (const float*)d_in[3];   // (V,E)
  const float* enc_att_W = (const float*)d_in[4];   const float* enc_att_b = (const float*)d_in[5];
  const float* dec_att_W = (const float*)d_in[6];   const float* dec_att_b = (const float*)d_in[7];
  const float* full_W    = (const float*)d_in[8];   const float* full_b    = (const float*)d_in[9];
  const float* init_h_W  = (const float*)d_in[10];  const float* init_h_b  = (const float*)d_in[11];
  const float* init_c_W  = (const float*)d_in[12];  const float* init_c_b  = (const float*)d_in[13];
  const float* f_beta_W  = (const float*)d_in[14];  const float* f_beta_b  = (const float*)d_in[15];
  const float* W_ih      = (const float*)d_in[16];  const float* b_ih      = (const float*)d_in[17];
  const float* W_hh      = (const float*)d_in[18];  const float* b_hh      = (const float*)d_in[19];
  const float* fc_W      = (const float*)d_in[20];  const float* fc_b      = (const float*)d_in[21];

  // --- output layout (tuple flattened, all as float) ---
  float* out          = (float*)d_out;
  float* out_preds    = out;                                   // B*TDEC*V
  float* out_caps     = out_preds  + (size_t)B_*TDEC_*V_;      // B*T
  float* out_declen   = out_caps   + (size_t)B_*T_;            // B
  float* out_alphas   = out_declen + (size_t)B_;               // B*TDEC*P
  float* out_sortind  = out_alphas + (size_t)B_*TDEC_*P_;      // B

  // --- workspace carve-out ---
  size_t off = 0;
  auto carve = [&](size_t bytes) -> void* {
    void* p = (char*)d_ws + off;
    off += (bytes + 255) & ~(size_t)255;
    return p;
  };
  int*   sort_ind  = (int*)  carve(B_*sizeof(int));
  int*   dec_len_i = (int*)  carve(B_*sizeof(int));
  int*   caps_s    = (int*)  carve((size_t)B_*T_*sizeof(int));
  int*   rowidx    = (int*)  carve((size_t)B_*P_*sizeof(int));
  float* mean_enc  = (float*)carve((size_t)B_*ENC_*sizeof(float));
  float* h         = (float*)carve((size_t)B_*D_*sizeof(float));
  float* c         = (float*)carve((size_t)B_*D_*sizeof(float));
  float* h_new     = (float*)carve((size_t)B_*D_*sizeof(float));
  float* att1      = (float*)carve((size_t)B_*P_*A_*sizeof(float));   // 25.7 MB
  float* att2      = (float*)carve((size_t)B_*A_*sizeof(float));
  float* e_buf     = (float*)carve((size_t)B_*P_*sizeof(float));
  float* awe       = (float*)carve((size_t)B_*ENC_*sizeof(float));
  float* gate      = (float*)carve((size_t)B_*ENC_*sizeof(float));
  float* x         = (float*)carve((size_t)B_*(E_+ENC_)*sizeof(float));
  float* gates     = (float*)carve((size_t)B_*4*D_*sizeof(float));
  (void)ws_size; (void)n_in; (void)in_sizes; (void)out_size;

  // --- sort / gather / pooling ---
  prep_kernel<<<1, B_, 0, stream>>>(cap_len, captions, sort_ind, dec_len_i, caps_s,
                                    out_caps, out_declen, out_sortind);
  rowidx_kernel<<<(B_*P_ + 255)/256, 256, 0, stream>>>(sort_ind, rowidx);
  mean_kernel<<<B_, 256, 0, stream>>>(enc, sort_ind, mean_enc);

  // h0 = tanh(mean_enc @ init_h_W + b); c0 = tanh(mean_enc @ init_c_W + b)
  wmma_gemm<1,false,false,false,false,4><<<dim3(8, 1), 128, 0, stream>>>(
      mean_enc, ENC_, init_h_W, D_, init_h_b, h, D_, D_, ENC_, nullptr, nullptr, 0);
  wmma_gemm<1,false,false,false,false,4><<<dim3(8, 1), 128, 0, stream>>>(
      mean_enc, ENC_, init_c_W, D_, init_c_b, c, D_, D_, ENC_, nullptr, nullptr, 0);

  // att1 = enc_sorted @ enc_att_W + b    (M=12544, N=512, K=2048, gathered rows)
  wmma_gemm<0,false,true,false,false,4><<<dim3(8, (B_*P_)/64), 128, 0, stream>>>(
      enc, ENC_, enc_att_W, A_, enc_att_b, att1, A_, A_, ENC_, rowidx, nullptr, 0);

  for (int t = 0; t < TDEC_; t++) {
    // att2 = h @ dec_att_W + b           (64x512 @ 512x512)
    wmma_gemm<0,false,false,false,false,4><<<dim3(8, 1), 128, 0, stream>>>(
        h, D_, dec_att_W, A_, dec_att_b, att2, A_, A_, D_, nullptr, nullptr, 0);
    // e = relu(att1 + att2) @ full_W + b
    e_kernel<<<(B_*P_)/8, 256, 0, stream>>>(att1, att2, full_W, full_b, e_buf);
    // alpha = softmax(e); awe = alpha @ enc ; masked alphas -> d_out
    softmax_awe_kernel<<<B_, 256, 0, stream>>>(e_buf, enc, sort_ind, dec_len_i, t,
                                               awe, out_alphas);
    // gate = sigmoid(h @ f_beta_W + b)   (64x512 @ 512x2048)
    wmma_gemm<2,false,false,false,false,4><<<dim3(32, 1), 128, 0, stream>>>(
        h, D_, f_beta_W, ENC_, f_beta_b, gate, ENC_, ENC_, D_, nullptr, nullptr, 0);
    // x = [emb_t ; gate*awe]
    build_x_kernel<<<B_, 256, 0, stream>>>(emb, caps_s, t, gate, awe, x);
    // gates = x @ W_ih^T + b_ih          (64x2560 @ 2560x2048)
    wmma_gemm<0,true,false,false,false,4><<<dim3(32, 1), 128, 0, stream>>>(
        x, E_+ENC_, W_ih, E_+ENC_, b_ih, gates, 4*D_, 4*D_, E_+ENC_, nullptr, nullptr, 0);
    // gates += h @ W_hh^T                (64x512 @ 512x2048, accumulate)
    wmma_gemm<0,true,false,true,false,4><<<dim3(32, 1), 128, 0, stream>>>(
        h, D_, W_hh, D_, nullptr, gates, 4*D_, 4*D_, D_, nullptr, nullptr, 0);
    // LSTM pointwise (+b_hh), masked state update
    lstm_kernel<<<B_, 256, 0, stream>>>(gates, b_hh, dec_len_i, t, h, c, h_new);
    // preds = h_new @ fc_W + fc_b, masked rows -> 0   (64x512 @ 512x10000)
    wmma_gemm<0,false,false,false,true,4><<<dim3((625 + 3)/4, 1), 128, 0, stream>>>(
        h_new, D_, fc_W, V_, fc_b, out_preds + (size_t)t*V_, TDEC_*V_, V_, D_,
        nullptr, dec_len_i, t);
  }
}